// SinkhornGAN_16595753632363
// MI455X (gfx1250) — compile-verified
//
#include <hip/hip_runtime.h>
#include <math.h>

// ---------------------------------------------------------------------------
// Sinkhorn loss on MI455X (gfx1250, wave32).
// M is never materialized: every kernel-matvec recomputes 16x16 tiles of
// -2*x*y^T with V_WMMA_F32_16X16X4_F32 and fuses the exp()/scale epilogue.
// B-tile fragments are double-buffered in registers so the 16-deep load
// clause for tile k+1 overlaps the 16 WMMAs + 8 v_exp_f32 of tile k.
// 4 waves per workgroup split the column loop (1024 waves per matvec).
// ---------------------------------------------------------------------------

typedef __attribute__((ext_vector_type(2))) float v2f;
typedef __attribute__((ext_vector_type(4))) float v4f;
typedef __attribute__((ext_vector_type(8))) float v8f;

static constexpr int   NPTS    = 4096;
static constexpr int   KD      = 64;
static constexpr float S_REG   = 10.0f;
static constexpr float INV_REG = 0.1f;
static constexpr float S_TAU   = 1000.0f;
static constexpr float INV_N   = 1.0f / 4096.0f;
static constexpr float S_EPS   = 1e-16f;

// ---------------------------------------------------------------------------
// Fused norms + init:
//   an2 = ||x_i||^2, bn2 = ||y_j||^2, ca = alpha - an2 = -an2,
//   cb = beta - bn2 = -bn2, u = v = 1/n.
// ---------------------------------------------------------------------------
__global__ __launch_bounds__(256) void sink_norminit_kernel(
    const float* __restrict__ x, const float* __restrict__ y,
    float* __restrict__ an2, float* __restrict__ bn2,
    float* __restrict__ ca, float* __restrict__ cb,
    float* __restrict__ u, float* __restrict__ v) {
  int idx = blockIdx.x * blockDim.x + threadIdx.x;  // 0..8191
  if (idx >= 2 * NPTS) return;
  const float* src = (idx < NPTS) ? x : y;
  int r = (idx < NPTS) ? idx : idx - NPTS;
  const v4f* p = (const v4f*)(src + (size_t)r * KD);
  float s = 0.0f;
#pragma unroll
  for (int q = 0; q < KD / 4; ++q) {
    v4f t = p[q];
    s += t.x * t.x + t.y * t.y + t.z * t.z + t.w * t.w;
  }
  if (idx < NPTS) {
    an2[r] = s; ca[r] = -s; u[r] = INV_N;
  } else {
    bn2[r] = s; cb[r] = -s; v[r] = INV_N;
  }
}

// ---------------------------------------------------------------------------
// Fused kernel-matvec:
//   out_i = INV_N / ( sum_j exp((coefA_i + coefB_j + 2<A_i,B_j>)/REG)*mul_j + eps )
// (coef* = potential - squared-norm, so the exp argument equals
//  (alpha_i + beta_j - M_ij)/REG with M = |A|^2 + |B|^2 - 2 A.B^T.)
//
// Workgroup = 128 threads (4 waves) owning 16 output rows; wave w handles
// column tiles j = w*16 + 64k. Per-wave WMMA accumulation with register
// double buffering of the B fragments, then LDS cross-wave combine.
//
// WMMA 32-bit A 16x4 layout: row = lane&15, K = 4*kt + 2*(lane>>4) + {0,1}.
// WMMA 32-bit C 16x16 layout: n = lane&15, m = t + 8*(lane>>4) in VGPR t.
// ---------------------------------------------------------------------------
__global__ __launch_bounds__(128) void sink_matvec_kernel(
    const float* __restrict__ A, const float* __restrict__ B,
    const float* __restrict__ coefA, const float* __restrict__ coefB,
    const float* __restrict__ mul, float* __restrict__ out) {
  __shared__ float lred[4][16];
  const int tid     = threadIdx.x;
  const int wave    = tid >> 5;
  const int lane    = tid & 31;
  const int l15     = lane & 15;
  const int half    = lane >> 4;
  const int rowbase = blockIdx.x * 16;

  // Preload the 16 A fragments for this 16-row strip.
  v2f afrag[16];
  const float* arow = A + (size_t)(rowbase + l15) * KD + half * 2;
#pragma unroll
  for (int kt = 0; kt < 16; ++kt)
    afrag[kt] = *(const v2f*)(arow + kt * 4);

  // Row coefficients for this lane's 8 C rows: m = t + half*8.
  float aa[8];
#pragma unroll
  for (int t = 0; t < 8; ++t)
    aa[t] = coefA[rowbase + half * 8 + t];

  float rsum[8] = {};

  auto load_btile = [&](int jj, v2f* bf) {
    const float* brow = B + (size_t)(jj + l15) * KD + half * 2;
#pragma unroll
    for (int kt = 0; kt < 16; ++kt)
      bf[kt] = *(const v2f*)(brow + kt * 4);
  };

  auto process = [&](int jj, const v2f* bf) {
    float cbv = coefB[jj + l15];
    float mv  = mul[jj + l15];
    v8f acc = {};
#pragma unroll
    for (int kt = 0; kt < 16; ++kt)
      acc = __builtin_amdgcn_wmma_f32_16x16x4_f32(
          false, afrag[kt], false, bf[kt], (short)0, acc, false, false);
#pragma unroll
    for (int t = 0; t < 8; ++t) {
      float arg = (aa[t] + cbv + 2.0f * acc[t]) * INV_REG;
      rsum[t] += __expf(arg) * mv;
    }
  };

  // Column loop: this wave handles tiles jbase + 64*k, double buffered.
  v2f b0[16], b1[16];
  const int jbase = wave * 16;
  load_btile(jbase, b0);
  for (int k = 0; k < 64; k += 2) {
    load_btile((jbase + (k + 1) * 64) & (NPTS - 1), b1);  // prefetch
    process(jbase + k * 64, b0);
    load_btile((jbase + (k + 2) * 64) & (NPTS - 1), b0);  // prefetch (wraps at end)
    process(jbase + (k + 1) * 64, b1);
  }

  // Reduce over the 16 lanes of each half (same rows, different columns).
#pragma unroll
  for (int t = 0; t < 8; ++t)
#pragma unroll
    for (int m = 1; m <= 8; m <<= 1)
      rsum[t] += __shfl_xor(rsum[t], m, 32);

  if (l15 == 0)
#pragma unroll
    for (int t = 0; t < 8; ++t)
      lred[wave][half * 8 + t] = rsum[t];
  __syncthreads();
  if (tid < 16) {
    float s = lred[0][tid] + lred[1][tid] + lred[2][tid] + lred[3][tid];
    out[rowbase + tid] = INV_N / (s + S_EPS);
  }
}

// ---------------------------------------------------------------------------
// Absorption (single workgroup): if max(|u|,|v|) > TAU, fold REG*log into the
// combined coefficients ca/cb and reset u, v to uniform.
// ---------------------------------------------------------------------------
__global__ __launch_bounds__(256) void sink_absorb_kernel(float* __restrict__ ca,
                                                          float* __restrict__ cb,
                                                          float* __restrict__ u,
                                                          float* __restrict__ v) {
  __shared__ float red[256];
  __shared__ int flag;
  const int tid = threadIdx.x;
  float mx = 0.0f;
  for (int i = tid; i < NPTS; i += 256) {
    mx = fmaxf(mx, fabsf(u[i]));
    mx = fmaxf(mx, fabsf(v[i]));
  }
  red[tid] = mx;
  __syncthreads();
  for (int s = 128; s > 0; s >>= 1) {
    if (tid < s) red[tid] = fmaxf(red[tid], red[tid + s]);
    __syncthreads();
  }
  if (tid == 0) flag = (red[0] > S_TAU) ? 1 : 0;
  __syncthreads();
  if (flag) {
    for (int i = tid; i < NPTS; i += 256) {
      ca[i] += S_REG * __logf(u[i]);
      cb[i] += S_REG * __logf(v[i]);
      u[i] = INV_N;
      v[i] = INV_N;
    }
  }
}

// ---------------------------------------------------------------------------
// Loss: sum_ij Gamma_ij * M_ij, Gamma = u_i*v_j*exp((ca_i+cb_j+2dot)/REG),
// M = an2_i + bn2_j - 2dot. Same WMMA tiling, 4-wave column split,
// deterministic per-block partial -> tree reduce (no float atomics).
// ---------------------------------------------------------------------------
__global__ __launch_bounds__(128) void sink_loss_kernel(
    const float* __restrict__ x, const float* __restrict__ y,
    const float* __restrict__ ca, const float* __restrict__ cb,
    const float* __restrict__ an2, const float* __restrict__ bn2,
    const float* __restrict__ u, const float* __restrict__ v,
    float* __restrict__ partials) {
  __shared__ float lred[4];
  const int tid     = threadIdx.x;
  const int wave    = tid >> 5;
  const int lane    = tid & 31;
  const int l15     = lane & 15;
  const int half    = lane >> 4;
  const int rowbase = blockIdx.x * 16;

  v2f afrag[16];
  const float* arow = x + (size_t)(rowbase + l15) * KD + half * 2;
#pragma unroll
  for (int kt = 0; kt < 16; ++kt)
    afrag[kt] = *(const v2f*)(arow + kt * 4);

  float cai[8], na[8], uu[8];
#pragma unroll
  for (int t = 0; t < 8; ++t) {
    int r = rowbase + half * 8 + t;
    cai[t] = ca[r];
    na[t]  = an2[r];
    uu[t]  = u[r];
  }

  float part = 0.0f;
  const int jbase = wave * 16;
  for (int k = 0; k < 64; ++k) {
    const int jj = jbase + k * 64;
    v2f bf[16];
    const float* brow = y + (size_t)(jj + l15) * KD + half * 2;
#pragma unroll
    for (int kt = 0; kt < 16; ++kt)
      bf[kt] = *(const v2f*)(brow + kt * 4);
    v8f acc = {};
#pragma unroll
    for (int kt = 0; kt < 16; ++kt)
      acc = __builtin_amdgcn_wmma_f32_16x16x4_f32(
          false, afrag[kt], false, bf[kt], (short)0, acc, false, false);
    float cbj = cb[jj + l15];
    float nb  = bn2[jj + l15];
    float vv  = v[jj + l15];
#pragma unroll
    for (int t = 0; t < 8; ++t) {
      float dot2 = 2.0f * acc[t];
      float M    = na[t] + nb - dot2;
      float arg  = (cai[t] + cbj + dot2) * INV_REG;
      part += __expf(arg) * uu[t] * vv * M;
    }
  }

#pragma unroll
  for (int m = 1; m <= 16; m <<= 1)
    part += __shfl_xor(part, m, 32);
  if (lane == 0) lred[wave] = part;
  __syncthreads();
  if (tid == 0)
    partials[blockIdx.x] = lred[0] + lred[1] + lred[2] + lred[3];
}

__global__ __launch_bounds__(256) void sink_reduce_kernel(const float* __restrict__ partials,
                                                          float* __restrict__ out) {
  __shared__ float s[256];
  const int tid = threadIdx.x;
  s[tid] = partials[tid];
  __syncthreads();
  for (int st = 128; st > 0; st >>= 1) {
    if (tid < st) s[tid] += s[tid + st];
    __syncthreads();
  }
  if (tid == 0) out[0] = s[0];
}

// ---------------------------------------------------------------------------
// Driver: norms+init, 50 x (col pass -> row pass -> absorb), loss, reduce.
// Workspace (floats): ca, cb, u, v, an2, bn2 (4096 each) + 256 partials.
// ---------------------------------------------------------------------------
extern "C" void kernel_launch(void* const* d_in, const int* in_sizes, int n_in,
                              void* d_out, int out_size, void* d_ws, size_t ws_size,
                              hipStream_t stream) {
  const float* x = (const float*)d_in[0];
  const float* y = (const float*)d_in[1];
  float* ws  = (float*)d_ws;
  float* ca  = ws;            // alpha - ||x||^2
  float* cb  = ws + 4096;     // beta  - ||y||^2
  float* u   = ws + 8192;
  float* v   = ws + 12288;
  float* an2 = ws + 16384;
  float* bn2 = ws + 20480;
  float* part = ws + 24576;   // 256 partials
  float* out = (float*)d_out;

  sink_norminit_kernel<<<32, 256, 0, stream>>>(x, y, an2, bn2, ca, cb, u, v);

  for (int it = 0; it < 50; ++it) {
    // v = b / (K^T u + eps): transposed roles (A=y/j, B=x/i, coef swapped)
    sink_matvec_kernel<<<NPTS / 16, 128, 0, stream>>>(y, x, cb, ca, u, v);
    // u = a / (K v + eps): natural roles, uses the freshly computed v
    sink_matvec_kernel<<<NPTS / 16, 128, 0, stream>>>(x, y, ca, cb, v, u);
    sink_absorb_kernel<<<1, 256, 0, stream>>>(ca, cb, u, v);
  }

  sink_loss_kernel<<<NPTS / 16, 128, 0, stream>>>(x, y, ca, cb, an2, bn2, u, v, part);
  sink_reduce_kernel<<<1, 256, 0, stream>>>(part, out);
}